// NodeNetwork_89644557402741
// MI455X (gfx1250) — compile-verified
//
#include <hip/hip_runtime.h>
#include <math.h>

// ---------------------------------------------------------------------------
// GNN node network for MI455X (gfx1250, wave32).
//   Phase 1: zero aggregation buffers (workspace).
//   Phase 2: edge scatter, 16 lanes/edge, global_atomic_add_f32 (L2-resident).
//   Phase 3: per-16-row-tile MLP using V_WMMA_F32_16X16X4_F32 (exact fp32),
//            activations staged in padded per-wave LDS, weights from L2/L0.
// ---------------------------------------------------------------------------

#define XS 132          // LDS row stride in floats (128 + 4 pad: bank-conflict free)
#define NWAVES 4        // waves per block -> 64 node rows per block

typedef __attribute__((ext_vector_type(2))) float v2f;
typedef __attribute__((ext_vector_type(8))) float v8f;

__device__ __forceinline__ void wsync() {
  // Per-wave DS ops are in-order in hardware; this only stops the compiler
  // from reordering or register-caching LDS values across cross-lane handoffs.
  __builtin_amdgcn_wave_barrier();
  asm volatile("" ::: "memory");
}

__global__ void zero_kernel(float* __restrict__ p, int n) {
  int i = blockIdx.x * blockDim.x + threadIdx.x;
  if (i < n) p[i] = 0.0f;
}

__global__ void scatter_kernel(const float* __restrict__ nodes,
                               const int*   __restrict__ edges,
                               const float* __restrict__ ew,
                               float* __restrict__ agg_in,
                               float* __restrict__ agg_out,
                               int E) {
  int t = blockIdx.x * blockDim.x + threadIdx.x;
  int e = t >> 4;          // edge index
  int d = t & 15;          // feature index (D=16)
  if (e >= E) return;
  int src = edges[2 * e];
  int dst = edges[2 * e + 1];
  float w = ew[e];
  // agg_in[dst]  += nodes[src] * w ; agg_out[src] += nodes[dst] * w
  atomicAdd(&agg_in [(size_t)dst * 16 + d], nodes[(size_t)src * 16 + d] * w);
  atomicAdd(&agg_out[(size_t)src * 16 + d], nodes[(size_t)dst * 16 + d] * w);
}

// One wave computes y[16 x (NT*16)] = LN(tanh-free) pipeline for one layer:
//   y = x @ W + bias, then LayerNorm + tanh, result back to LDS (or to gout).
template <int NT>
__device__ __forceinline__ void layer(float* xb,
                                      const float* __restrict__ W,
                                      const float* __restrict__ bias,
                                      const float* __restrict__ g,
                                      const float* __restrict__ be,
                                      int K, int lane,
                                      float* __restrict__ gout,
                                      int tilebase, int N) {
  const int Nout = NT * 16;
  const int half = lane >> 4;   // which 16-lane half of the wave
  const int m    = lane & 15;   // A row for this lane; also B/C/D column

  v8f acc[NT] = {};

  for (int kb = 0; kb < K; kb += 4) {
    int k = kb + 2 * half;      // ISA layout: VGPR j holds k = kbase + j + 2*half
    v2f a;
    a.x = xb[m * XS + k];       // merged to ds_load_b64 (8B aligned: XS,k even)
    a.y = xb[m * XS + k + 1];
    const float* wp = W + (size_t)k * Nout + m;   // column m of W row k
#pragma unroll
    for (int t = 0; t < NT; ++t) {
      v2f b;
      b.x = wp[t * 16];         // W[k  ][t*16 + m]  (64B coalesced per 16 lanes)
      b.y = wp[t * 16 + Nout];  // W[k+1][t*16 + m]
      acc[t] = __builtin_amdgcn_wmma_f32_16x16x4_f32(
          /*neg_a=*/false, a, /*neg_b=*/false, b,
          /*c_mod=*/(short)0, acc[t], /*reuse_a=*/false, /*reuse_b=*/false);
    }
  }
  wsync();

  // Spill C/D tiles (+bias) to LDS.  Layout: VGPR v -> row v + 8*half, col = m.
#pragma unroll
  for (int t = 0; t < NT; ++t) {
    float bv = bias[t * 16 + m];
#pragma unroll
    for (int v = 0; v < 8; ++v) {
      xb[(v + 8 * half) * XS + t * 16 + m] = acc[t][v] + bv;
    }
  }
  wsync();

  // LayerNorm + tanh: lanes 0..15 each own one row of Nout values.
  if (lane < 16) {
    float s = 0.0f, sq = 0.0f;
    for (int c = 0; c < Nout; ++c) {
      float v = xb[lane * XS + c];
      s += v; sq += v * v;
    }
    float mean = s / (float)Nout;
    float var  = sq / (float)Nout - mean * mean;
    float rs   = rsqrtf(var + 1e-5f);
    if (gout) {
      int grow = tilebase + lane;
      if (grow < N) {
        for (int c = 0; c < Nout; ++c) {
          float v = (xb[lane * XS + c] - mean) * rs * g[c] + be[c];
          gout[(size_t)grow * Nout + c] = tanhf(v);
        }
      }
    } else {
      for (int c = 0; c < Nout; ++c) {
        float v = (xb[lane * XS + c] - mean) * rs * g[c] + be[c];
        xb[lane * XS + c] = tanhf(v);
      }
    }
  }
  wsync();
}

__global__ void __launch_bounds__(NWAVES * 32) mlp_kernel(
    const float* __restrict__ nodes,
    const float* __restrict__ agg_in,
    const float* __restrict__ agg_out,
    const float* __restrict__ W1, const float* __restrict__ b1,
    const float* __restrict__ g1, const float* __restrict__ be1,
    const float* __restrict__ W2, const float* __restrict__ b2,
    const float* __restrict__ g2, const float* __restrict__ be2,
    const float* __restrict__ W3, const float* __restrict__ b3,
    const float* __restrict__ g3, const float* __restrict__ be3,
    const float* __restrict__ W4, const float* __restrict__ b4,
    const float* __restrict__ g4, const float* __restrict__ be4,
    float* __restrict__ out, int N) {
  __shared__ float xbuf[NWAVES][16 * XS];
  int wave = threadIdx.x >> 5;
  int lane = threadIdx.x & 31;
  int tilebase = (blockIdx.x * NWAVES + wave) * 16;
  if (tilebase >= N) return;          // wave-uniform exit: EXEC stays all-ones
  float* xb = xbuf[wave];

  // Stage x0 = [agg_in | agg_out | nodes] : 16 rows x 48 cols (zero-pad tail).
  for (int idx = lane; idx < 16 * 48; idx += 32) {
    int r = idx / 48;
    int c = idx - r * 48;
    int grow = tilebase + r;
    float v = 0.0f;
    if (grow < N) {
      if (c < 16)       v = agg_in [(size_t)grow * 16 + c];
      else if (c < 32)  v = agg_out[(size_t)grow * 16 + (c - 16)];
      else              v = nodes  [(size_t)grow * 16 + (c - 32)];
    }
    xb[r * XS + c] = v;
  }
  wsync();

  layer<8>(xb, W1, b1, g1, be1,  48, lane, nullptr, 0, N);
  layer<8>(xb, W2, b2, g2, be2, 128, lane, nullptr, 0, N);
  layer<8>(xb, W3, b3, g3, be3, 128, lane, nullptr, 0, N);
  layer<1>(xb, W4, b4, g4, be4, 128, lane, out, tilebase, N);
}

extern "C" void kernel_launch(void* const* d_in, const int* in_sizes, int n_in,
                              void* d_out, int out_size, void* d_ws, size_t ws_size,
                              hipStream_t stream) {
  const float* nodes = (const float*)d_in[0];
  const int*   edges = (const int*)  d_in[1];
  const float* ew    = (const float*)d_in[2];
  const float* W1 = (const float*)d_in[3];
  const float* b1 = (const float*)d_in[4];
  const float* g1 = (const float*)d_in[5];
  const float* be1= (const float*)d_in[6];
  const float* W2 = (const float*)d_in[7];
  const float* b2 = (const float*)d_in[8];
  const float* g2 = (const float*)d_in[9];
  const float* be2= (const float*)d_in[10];
  const float* W3 = (const float*)d_in[11];
  const float* b3 = (const float*)d_in[12];
  const float* g3 = (const float*)d_in[13];
  const float* be3= (const float*)d_in[14];
  const float* W4 = (const float*)d_in[15];
  const float* b4 = (const float*)d_in[16];
  const float* g4 = (const float*)d_in[17];
  const float* be4= (const float*)d_in[18];

  int N = in_sizes[0] / 16;   // D = 16
  int E = in_sizes[2];

  float* agg_in  = (float*)d_ws;
  float* agg_out = agg_in + (size_t)N * 16;

  int zn = N * 32;            // both aggregation buffers
  zero_kernel<<<(zn + 255) / 256, 256, 0, stream>>>(agg_in, zn);

  long long st = (long long)E * 16;
  scatter_kernel<<<(int)((st + 255) / 256), 256, 0, stream>>>(
      nodes, edges, ew, agg_in, agg_out, E);

  int nblocks = (N + NWAVES * 16 - 1) / (NWAVES * 16);
  mlp_kernel<<<nblocks, NWAVES * 32, 0, stream>>>(
      nodes, agg_in, agg_out,
      W1, b1, g1, be1, W2, b2, g2, be2,
      W3, b3, g3, be3, W4, b4, g4, be4,
      (float*)d_out, N);
}